// KANPointNet_11089605558326
// MI455X (gfx1250) — compile-verified
//
#include <hip/hip_runtime.h>
#include <stdint.h>

// KAN PointNet, MI455X (gfx1250).
//
// Dead-code analysis of the reference: final output (kan10, 40 ch) depends only
// on the chain kan1->kan2->kan6->kan7->kan8->kan9->kan10 restricted to
// channels 0..39 (sel = j % Cin is identity for every post-concat layer, and
// kan6 ch<64 reads the *local* half of the concat). The global-pool branch is
// dead. Remaining work: 8*8192 points x 40 ch x 7 spline evals, ~11 MB of
// mandatory HBM traffic -> fully fused single kernel, memory-trivial.
//
// CDNA5 specifics used:
//  - global_load_async_to_lds_b32 + s_wait_asynccnt to stage knot tables
//  - ds_load_b128 gathers of transposed coef quads (16B-aligned)
//  - wave32 / 8-wave blocks

#define NN    8192
#define NB    8
#define NCH   40
#define NL    7
#define TPB   256
#define KNOTS_PER_LAYER (NCH * 6)    // 240 dwords
#define COEFS_PER_LAYER (NCH * 20)   // 800 dwords

__device__ __forceinline__ void async_ld_lds_b32(void* ldsDst, const float* gBase,
                                                 uint32_t dwordIdx) {
#if defined(__HIP_DEVICE_COMPILE__)
    // Low 32 bits of a flat shared pointer == LDS byte offset (aperture layout).
    uint32_t lds = (uint32_t)(uintptr_t)ldsDst;
    uint32_t off = dwordIdx * 4u;
    asm volatile("global_load_async_to_lds_b32 %0, %1, %2"
                 :
                 : "v"(lds), "v"(off), "s"(gBase)
                 : "memory");
#endif
}

__device__ __forceinline__ void wait_asynccnt0() {
#if defined(__HIP_DEVICE_COMPILE__)
    asm volatile("s_wait_asynccnt 0" ::: "memory");
#endif
}

// One spline eval. tk: 6 knots (LDS). cf: 5 intervals * 4 coefs, [i][p] layout
// (LDS, 16B aligned) so the gather is a single ds_load_b128.
__device__ __forceinline__ float spline_eval(const float* __restrict__ tk,
                                             const float* __restrict__ cf,
                                             float x) {
    // searchsorted(t, x, 'right') - 1, clipped to [0, 4], for sorted knots:
    int i = (x >= tk[1]) + (x >= tk[2]) + (x >= tk[3]) + (x >= tk[4]);
    float dx = x - tk[i];
    const float4 c = *reinterpret_cast<const float4*>(cf + (i << 2));
    return ((c.x * dx + c.y) * dx + c.z) * dx + c.w;   // c0*dx^3+c1*dx^2+c2*dx+c3
}

__global__ __launch_bounds__(TPB) void kan_fused_kernel(
    const float* __restrict__ x,      // [8,3,8192]
    float* __restrict__ out,          // [8,40,8192]
    const float* __restrict__ kn0, const float* __restrict__ cf0,   // kan1
    const float* __restrict__ kn1, const float* __restrict__ cf1,   // kan2
    const float* __restrict__ kn2, const float* __restrict__ cf2,   // kan6
    const float* __restrict__ kn3, const float* __restrict__ cf3,   // kan7
    const float* __restrict__ kn4, const float* __restrict__ cf4,   // kan8
    const float* __restrict__ kn5, const float* __restrict__ cf5,   // kan9
    const float* __restrict__ kn6, const float* __restrict__ cf6)   // kan10
{
    // LDS tables: knots [l][ch][k0..5], coefs transposed [l][ch][i][p]
    __shared__ __align__(16) float sKnots[NL * KNOTS_PER_LAYER];   // 6720 B
    __shared__ __align__(16) float sCoefs[NL * COEFS_PER_LAYER];   // 22400 B

    const float* kn[NL] = {kn0, kn1, kn2, kn3, kn4, kn5, kn6};
    const float* cf[NL] = {cf0, cf1, cf2, cf3, cf4, cf5, cf6};

    const int t = threadIdx.x;

    // --- Stage knots via CDNA5 async global->LDS copies (no transpose needed).
    if (t < KNOTS_PER_LAYER) {
        #pragma unroll
        for (int l = 0; l < NL; ++l) {
            async_ld_lds_b32(&sKnots[l * KNOTS_PER_LAYER + t], kn[l], (uint32_t)t);
        }
    }

    // --- Stage coefs with [p][i] -> [i][p] transpose (regular load + ds_store),
    //     so each eval gathers one contiguous float4.
    #pragma unroll
    for (int l = 0; l < NL; ++l) {
        const float* g = cf[l];
        for (int j = t; j < COEFS_PER_LAYER; j += TPB) {
            int ch = j / 20;
            int r  = j - ch * 20;
            int p  = r / 5;
            int i  = r - p * 5;
            sCoefs[(l * NCH + ch) * 20 + i * 4 + p] = g[j];
        }
    }

    wait_asynccnt0();      // async LDS writes from this wave complete
    __syncthreads();       // all waves' LDS writes visible

    // --- Fused 7-layer chain, one point per thread.
    const int gid = blockIdx.x * TPB + t;       // 0 .. 65535
    const int b   = gid >> 13;                  // gid / 8192
    const int n   = gid & (NN - 1);             // gid % 8192

    const float* xb = x + ((size_t)b * 3) * NN + n;
    const float x0 = xb[0];
    const float x1 = xb[NN];
    const float x2 = xb[2 * NN];

    float* ob = out + ((size_t)b * NCH) * NN + n;

    int c = 0;  // ch % 3, maintained incrementally
    #pragma unroll 2
    for (int ch = 0; ch < NCH; ++ch) {
        float v = (c == 0) ? x0 : ((c == 1) ? x1 : x2);
        c = (c == 2) ? 0 : c + 1;

        #pragma unroll
        for (int l = 0; l < NL; ++l) {
            const float* tk = &sKnots[(l * NCH + ch) * 6];
            const float* cc = &sCoefs[(l * NCH + ch) * 20];
            v = spline_eval(tk, cc, v);
            if (l < NL - 1) v = fmaxf(v, 0.0f);   // ReLU between layers, not after kan10
        }
        ob[(size_t)ch * NN] = v;
    }
}

extern "C" void kernel_launch(void* const* d_in, const int* in_sizes, int n_in,
                              void* d_out, int out_size, void* d_ws, size_t ws_size,
                              hipStream_t stream) {
    (void)in_sizes; (void)n_in; (void)d_ws; (void)ws_size; (void)out_size;
    const float* x = (const float*)d_in[0];
    // setup_inputs order: x, knots1, coefs1, ..., knots10, coefs10
    // Live layers: kan1, kan2, kan6, kan7, kan8, kan9, kan10
    auto KN = [&](int li) { return (const float*)d_in[2 * li - 1]; };
    auto CF = [&](int li) { return (const float*)d_in[2 * li]; };

    float* out = (float*)d_out;
    dim3 grid((NB * NN) / TPB);   // 256 blocks
    dim3 block(TPB);
    kan_fused_kernel<<<grid, block, 0, stream>>>(
        x, out,
        KN(1), CF(1),
        KN(2), CF(2),
        KN(6), CF(6),
        KN(7), CF(7),
        KN(8), CF(8),
        KN(9), CF(9),
        KN(10), CF(10));
}